// MovingAvgNorm_44607530336438
// MI455X (gfx1250) — compile-verified
//
#include <hip/hip_runtime.h>
#include <cstdint>

// MovingAvgNorm: y = (x - mean_k(x)) / (std_k(x) + eps), k=100, reflect pad,
// x: (32, 16384, 80) f32. Pure bandwidth problem (~368MB @ 23.3TB/s ~ 16us).
// CDNA5 strategy: stage a (CH+99)-row time tile in LDS via async global->LDS
// DMA (global_load_async_to_lds_b128, ASYNCcnt), then serve the ~100x data
// reuse from LDS. All data movement is 128-bit: b128 async loads, b128 LDS
// reads, b128 global stores (each lane owns 4 consecutive channels).

#define B_    32
#define T_    16384
#define D_    80
#define K_    100
#define CH    512                  // outputs per block along time
#define ROWS  (CH + K_ - 1)        // 611 padded-time rows staged in LDS
#define SEGS  (D_ / 4)             // 20 float4 segments per row
#define NTHREADS 320               // 16 strips x 20 lane-groups (10 wave32)
#define STRIP 32                   // outputs per thread (x4 channels)

typedef float v4f __attribute__((ext_vector_type(4)));

__device__ __forceinline__ int reflect_q(int q) {
  // single reflection suffices since K << T
  if (q < 0) q = -q;
  else if (q >= T_) q = 2 * T_ - 2 - q;
  return q;
}

__global__ __launch_bounds__(NTHREADS)
void MovingAvgNorm_kernel(const float* __restrict__ x, float* __restrict__ out) {
  __shared__ __align__(16) v4f tile[ROWS * SEGS];   // 611*80*4 = 195,520 B

  const int b   = blockIdx.x >> 5;          // blockIdx.x / 32
  const int t0  = (blockIdx.x & 31) * CH;   // chunk start time
  const int tid = threadIdx.x;
  const int seg = tid % SEGS;               // which 16B segment of a row
  const int grp = tid / SEGS;               // 0..15

  // ---- Phase 1: async DMA the padded tile into LDS (reflect applied here) ----
  // 611 rows x 20 segments of 16B; fixed seg per thread, stride over rows.
  const v4f* xv = (const v4f*)x;
  for (int r = grp; r < ROWS; r += NTHREADS / SEGS) {
    const int q = reflect_q(t0 + r - (K_ / 2));
    const unsigned long long gaddr =
        (unsigned long long)(uintptr_t)(xv + (size_t)(b * T_ + q) * SEGS + seg);
    const uint32_t ldsaddr =
        (uint32_t)(uintptr_t)(&tile[r * SEGS + seg]);  // low 32b of generic = LDS offset
    asm volatile("global_load_async_to_lds_b128 %0, %1, off"
                 :: "v"(ldsaddr), "v"(gaddr) : "memory");
  }
#if defined(__has_builtin) && __has_builtin(__builtin_amdgcn_s_wait_asynccnt)
  __builtin_amdgcn_s_wait_asynccnt(0);
#else
  asm volatile("s_wait_asynccnt 0x0" ::: "memory");
#endif
  __syncthreads();

  // ---- Phase 2: sliding-window mean/std, 4 channels per lane, all b128 ----
  const int base = grp * STRIP;             // first tile row of this strip

  v4f s1 = (v4f)0.0f, s2 = (v4f)0.0f;
#pragma unroll 4
  for (int j = 0; j < K_; ++j) {            // ramp: first window of the strip
    const v4f v = tile[(base + j) * SEGS + seg];
    s1 += v;
    s2 += v * v;
  }

  const float inv_k   = 1.0f / (float)K_;
  const float inv_km1 = 1.0f / (float)(K_ - 1);
  v4f* const outv = (v4f*)out;
#pragma unroll 2
  for (int u = 0; u < STRIP; ++u) {
    const v4f xc  = tile[(base + u + K_ / 2) * SEGS + seg];  // center x[t]
    const v4f m   = s1 * inv_k;
    const v4f var = (s2 - s1 * s1 * inv_k) * inv_km1;        // unbiased
    v4f sd;
    sd.x = sqrtf(fmaxf(var.x, 0.0f));
    sd.y = sqrtf(fmaxf(var.y, 0.0f));
    sd.z = sqrtf(fmaxf(var.z, 0.0f));
    sd.w = sqrtf(fmaxf(var.w, 0.0f));
    const int t = t0 + base + u;
    outv[(size_t)(b * T_ + t) * SEGS + seg] = (xc - m) / (sd + 1e-12f);
    if (u < STRIP - 1) {                    // slide window by one time step
      const v4f vo = tile[(base + u) * SEGS + seg];
      const v4f vn = tile[(base + u + K_) * SEGS + seg];
      s1 += vn - vo;
      s2 += vn * vn - vo * vo;
    }
  }
}

extern "C" void kernel_launch(void* const* d_in, const int* in_sizes, int n_in,
                              void* d_out, int out_size, void* d_ws, size_t ws_size,
                              hipStream_t stream) {
  (void)in_sizes; (void)n_in; (void)d_ws; (void)ws_size; (void)out_size;
  const float* x = (const float*)d_in[0];
  float* out     = (float*)d_out;
  const dim3 grid(B_ * (T_ / CH));   // 32 batches * 32 chunks = 1024 blocks
  MovingAvgNorm_kernel<<<grid, NTHREADS, 0, stream>>>(x, out);
}